// GraphNetwork_63479616635203
// MI455X (gfx1250) — compile-verified
//
#include <hip/hip_runtime.h>

// GAT-style 2-layer graph network for MI455X (gfx1250, wave32).
// Heavy math (state@Wf and att@Wh GEMMs) runs on V_WMMA_F32_16X16X4_F32.
// Softmax over rows is restructured into per-column (max, 1/sum) stats so the
// attention matmul is a pure LDS-tiled WMMA GEMM with exp() computed on the fly.

typedef __attribute__((ext_vector_type(2))) float v2f;
typedef __attribute__((ext_vector_type(8))) float v8f;

constexpr int B = 4, N = 2048, I = 32, H = 64, O = 32, E = 2, HD = 2;
constexpr int NC  = E * HD;      // 4 (e-major, h-minor) combos
constexpr int BN  = B * N;       // 8192
constexpr int BNH = BN * H;      // 524288
constexpr int D1  = H * HD * E;  // 256
constexpr int LDA = 65;          // LDS pitch for A tiles (bank-conflict-free frag reads)
constexpr int LDB = 72;          // LDS pitch for B tiles (lane vs lane+16 different banks)

static __device__ __forceinline__ float lrelu(float x) { return x > 0.f ? x : 0.2f * x; }

// One 64-wide K-chunk of D += A(16x64) * B(64x64) per wave, 4 col-tiles of 16.
// A-frag layout (ISA 7.12.2, 32-bit A 16x4): lanes 0-15 = rows, VGPR0 holds
// K={0|2} by lane-half, VGPR1 holds K={1|3}.  B assumed mirrored (row-striped).
static __device__ __forceinline__ void wmma_kchunk(const float* As, const float* Bs,
                                                   v8f acc[4], int wave, int lane) {
  const int rm = lane & 15, kh = lane >> 4;
  const float* arow = As + (wave * 16 + rm) * LDA;
#pragma unroll
  for (int kk = 0; kk < 64; kk += 4) {
    v2f a;
    a.x = arow[kk + 2 * kh];
    a.y = arow[kk + 2 * kh + 1];
#pragma unroll
    for (int ct = 0; ct < 4; ++ct) {
      v2f bf;
      bf.x = Bs[(kk + 2 * kh) * LDB + ct * 16 + rm];
      bf.y = Bs[(kk + 2 * kh + 1) * LDB + ct * 16 + rm];
      acc[ct] = __builtin_amdgcn_wmma_f32_16x16x4_f32(false, a, false, bf,
                                                      (short)0, acc[ct], false, false);
    }
  }
}

// state0 = nodes @ W_emb + b_emb   (tiny: 33 MFLOP)
__global__ void gat_embed(const float* __restrict__ nodes, const float* __restrict__ W_emb,
                          const float* __restrict__ b_emb, float* __restrict__ state0) {
  int gid = blockIdx.x * blockDim.x + threadIdx.x;
  if (gid >= BN * H) return;
  int row = gid >> 6, col = gid & 63;
  float acc = b_emb[col];
  const float* nr = nodes + row * I;
#pragma unroll
  for (int i = 0; i < I; ++i) acc = fmaf(nr[i], W_emb[i * H + col], acc);
  state0[gid] = acc;
}

// Wh[combo] = state (BN x DIN) @ Wf[combo] (DIN x 64), WMMA-tiled via LDS.
template <int DIN>
__launch_bounds__(128)
__global__ void gat_wh(const float* __restrict__ state, const float* __restrict__ Wf,
                       float* __restrict__ Wh) {
  __shared__ float As[64 * LDA];
  __shared__ float Bs[64 * LDB];
  const int combo = blockIdx.y;
  const int row0  = blockIdx.x * 64;
  const int t = threadIdx.x, wave = t >> 5, lane = t & 31;
  const int rm = lane & 15, kh = lane >> 4;
  const int colf = t & 63, rhalf = t >> 6;
  v8f acc[4] = {};
  for (int k0 = 0; k0 < DIN; k0 += 64) {
#pragma unroll
    for (int r2 = 0; r2 < 64; r2 += 2) {
      int r = r2 + rhalf;
      As[r * LDA + colf] = state[(row0 + r) * DIN + k0 + colf];
      Bs[r * LDB + colf] = Wf[combo * (DIN * H) + (k0 + r) * H + colf];
    }
    __syncthreads();
    wmma_kchunk(As, Bs, acc, wave, lane);
    __syncthreads();
  }
  float* out = Wh + combo * BNH;
#pragma unroll
  for (int ct = 0; ct < 4; ++ct)
#pragma unroll
    for (int j = 0; j < 8; ++j) {
      int r = wave * 16 + j + 8 * kh;             // C/D layout: VGPR j -> M=j / j+8
      out[(row0 + r) * H + ct * 16 + rm] = acc[ct][j];
    }
}

// a1/a2 row scores: one wave32 per row, butterfly reduce.
__global__ void gat_a12(const float* __restrict__ Wh,
                        const float* __restrict__ w1, const float* __restrict__ b1,
                        const float* __restrict__ w2, const float* __restrict__ b2,
                        float* __restrict__ a1, float* __restrict__ a2) {
  int gtid = blockIdx.x * blockDim.x + threadIdx.x;
  int rowIdx = gtid >> 5, lane = gtid & 31;
  if (rowIdx >= NC * BN) return;
  int combo = rowIdx / BN;
  const float* wr  = Wh + rowIdx * H;
  const float* w1r = w1 + combo * H;
  const float* w2r = w2 + combo * H;
  float x0 = wr[lane], x1 = wr[32 + lane];
  float v1 = x0 * w1r[lane] + x1 * w1r[32 + lane];
  float v2 = x0 * w2r[lane] + x1 * w2r[32 + lane];
#pragma unroll
  for (int m = 16; m > 0; m >>= 1) { v1 += __shfl_xor(v1, m); v2 += __shfl_xor(v2, m); }
  if (lane == 0) {
    a1[rowIdx] = v1 + b1[combo];
    a2[rowIdx] = v2 + b2[combo];
  }
}

// Per-column softmax stats over rows i, all 4 (e,h) combos per edges read.
// Reads edges as float2 (both edge types) -> each element touched once/pass.
__global__ void gat_colstats(const float* __restrict__ edges,
                             const float* __restrict__ a1, const float* __restrict__ a2,
                             float* __restrict__ colm, float* __restrict__ colsinv) {
  __shared__ float a1s[NC][N];                      // 32 KB
  const int b = blockIdx.y;
  const int k = blockIdx.x * blockDim.x + threadIdx.x;
  for (int idx = threadIdx.x; idx < NC * N; idx += blockDim.x) {
    int c = idx >> 11, i = idx & (N - 1);
    a1s[c][i] = a1[c * BN + b * N + i];
  }
  __syncthreads();
  float a2r[NC], m[NC], s[NC];
#pragma unroll
  for (int c = 0; c < NC; ++c) { a2r[c] = a2[c * BN + b * N + k]; m[c] = -3.0e38f; }
  const float2* erow = (const float2*)edges;        // [(b*N+i)*N + k] -> (e0,e1)
  for (int i = 0; i < N; ++i) {
    float2 eb = erow[(b * N + i) * N + k];
#pragma unroll
    for (int c = 0; c < NC; ++c) {
      float ev = (c & 2) ? eb.y : eb.x;
      float l = lrelu(a1s[c][i] + a2r[c]) + ev;
      m[c] = fmaxf(m[c], l);
    }
  }
#pragma unroll
  for (int c = 0; c < NC; ++c) s[c] = 0.f;
  for (int i = 0; i < N; ++i) {
    float2 eb = erow[(b * N + i) * N + k];
#pragma unroll
    for (int c = 0; c < NC; ++c) {
      float ev = (c & 2) ? eb.y : eb.x;
      float l = lrelu(a1s[c][i] + a2r[c]) + ev;
      s[c] += __expf(l - m[c]);
    }
  }
#pragma unroll
  for (int c = 0; c < NC; ++c) {
    colm[c * BN + b * N + k]   = m[c];
    colsinv[c * BN + b * N + k] = 1.0f / s[c];
  }
}

// msg = P @ (Wh * sinv[k]) with P[i,k] = exp(logit - m[k]) built in LDS.
// MODE 0: layer-0 -> concat into state1 with bias0[h]; MODE 1: raw msgbuf.
template <int MODE>
__launch_bounds__(128)
__global__ void gat_msg(const float* __restrict__ edges,
                        const float* __restrict__ a1, const float* __restrict__ a2,
                        const float* __restrict__ colm, const float* __restrict__ colsinv,
                        const float* __restrict__ Wh, const float* __restrict__ bias,
                        float* __restrict__ out) {
  __shared__ float Ps[64 * LDA];
  __shared__ float Bs[64 * LDB];
  __shared__ float a1s[64];
  const int combo = blockIdx.z, e = combo >> 1, h = combo & 1;
  const int b = blockIdx.y;
  const int i0 = blockIdx.x * 64;
  const int t = threadIdx.x, wave = t >> 5, lane = t & 31;
  const int rm = lane & 15, kh = lane >> 4;
  const int colf = t & 63, rhalf = t >> 6;
  const int cb = combo * BN + b * N;
  if (t < 64) a1s[t] = a1[cb + i0 + t];
  const float* whc = Wh + combo * BNH + b * N * H;
  v8f acc[4] = {};
  for (int k0 = 0; k0 < N; k0 += 64) {
    const float a2r = a2[cb + k0 + colf];           // this thread's column
    const float mr  = colm[cb + k0 + colf];
    __syncthreads();                                // a1s ready / LDS reusable
#pragma unroll
    for (int r2 = 0; r2 < 64; r2 += 2) {
      int r = r2 + rhalf;
      float ev = edges[(((b * N + i0 + r) * N + k0 + colf) << 1) + e];
      float l  = lrelu(a1s[r] + a2r) + ev;
      Ps[r * LDA + colf] = __expf(l - mr);
      Bs[r * LDB + colf] = whc[(k0 + r) * H + colf] * colsinv[cb + k0 + r];
    }
    __syncthreads();
    wmma_kchunk(Ps, Bs, acc, wave, lane);
  }
#pragma unroll
  for (int ct = 0; ct < 4; ++ct)
#pragma unroll
    for (int j = 0; j < 8; ++j) {
      int r = wave * 16 + j + 8 * kh;
      int c = ct * 16 + rm;
      float v = acc[ct][j];
      if (MODE == 0)
        out[(b * N + i0 + r) * D1 + combo * H + c] = v + bias[h * H + c];
      else
        out[combo * BNH + (b * N + i0 + r) * H + c] = v;
    }
}

// Layer-1 tail: state2 = mean over (e,h) of elu(msg + bias1[h]).
__global__ void gat_elu_mean(const float* __restrict__ msgbuf, const float* __restrict__ bias,
                             float* __restrict__ state2) {
  int gid = blockIdx.x * blockDim.x + threadIdx.x;
  if (gid >= BNH) return;
  int d = gid & 63;
  float acc = 0.f;
#pragma unroll
  for (int c = 0; c < NC; ++c) {
    float v = msgbuf[c * BNH + gid] + bias[(c & 1) * H + d];
    acc += (v > 0.f) ? v : (__expf(v) - 1.f);
  }
  state2[gid] = acc * 0.25f;
}

// out = mean_n(state2) @ W_out + b_out  (mean commutes with the linear map)
__global__ void gat_final(const float* __restrict__ state2, const float* __restrict__ W_out,
                          const float* __restrict__ b_out, float* __restrict__ outp) {
  __shared__ float rmean[B * H];
  int t = threadIdx.x;
  {
    int b = t >> 6, d = t & 63;
    float acc = 0.f;
    const float* p = state2 + b * N * H + d;
    for (int n = 0; n < N; ++n) acc += p[n * H];
    rmean[t] = acc * (1.0f / N);
  }
  __syncthreads();
  if (t < B * O) {
    int b = t >> 5, o = t & 31;
    float s = b_out[o];
#pragma unroll
    for (int hh = 0; hh < H; ++hh) s = fmaf(rmean[b * H + hh], W_out[hh * O + o], s);
    outp[t] = s;
  }
}

extern "C" void kernel_launch(void* const* d_in, const int* in_sizes, int n_in,
                              void* d_out, int out_size, void* d_ws, size_t ws_size,
                              hipStream_t stream) {
  (void)in_sizes; (void)n_in; (void)out_size; (void)ws_size;
  const float* nodes = (const float*)d_in[0];
  const float* edges = (const float*)d_in[1];
  const float* W_emb = (const float*)d_in[2];
  const float* b_emb = (const float*)d_in[3];
  const float* Wf0   = (const float*)d_in[4];
  const float* w1_0  = (const float*)d_in[5];
  const float* b1_0  = (const float*)d_in[6];
  const float* w2_0  = (const float*)d_in[7];
  const float* b2_0  = (const float*)d_in[8];
  const float* bias0 = (const float*)d_in[9];
  const float* Wf1   = (const float*)d_in[10];
  const float* w1_1  = (const float*)d_in[11];
  const float* b1_1  = (const float*)d_in[12];
  const float* w2_1  = (const float*)d_in[13];
  const float* b2_1  = (const float*)d_in[14];
  const float* bias1 = (const float*)d_in[15];
  const float* W_out = (const float*)d_in[16];
  const float* b_out = (const float*)d_in[17];

  // Workspace (~18.5 MB, all float32), with safe aliasing:
  float* ws     = (float*)d_ws;
  float* state0 = ws;                        // BNH
  float* Wh     = state0 + BNH;              // NC*BNH
  float* a1     = Wh + (size_t)NC * BNH;     // NC*BN
  float* a2     = a1 + NC * BN;              // NC*BN
  float* colm   = a2 + NC * BN;              // NC*BN
  float* sinv   = colm + NC * BN;            // NC*BN
  float* state1 = sinv + NC * BN;            // BN*D1
  float* msgbuf = state1;                    // alias: state1 dead after layer-1 Wh
  float* state2 = state0;                    // alias: state0 dead after layer-0 Wh

  gat_embed<<<dim3((BN * H) / 256), 256, 0, stream>>>(nodes, W_emb, b_emb, state0);

  // ---- layer 0 ----
  gat_wh<H><<<dim3(BN / 64, NC), 128, 0, stream>>>(state0, Wf0, Wh);
  gat_a12<<<dim3((NC * BN * 32) / 256), 256, 0, stream>>>(Wh, w1_0, b1_0, w2_0, b2_0, a1, a2);
  gat_colstats<<<dim3(N / 256, B), 256, 0, stream>>>(edges, a1, a2, colm, sinv);
  gat_msg<0><<<dim3(N / 64, B, NC), 128, 0, stream>>>(edges, a1, a2, colm, sinv, Wh, bias0, state1);

  // ---- layer 1 ----
  gat_wh<D1><<<dim3(BN / 64, NC), 128, 0, stream>>>(state1, Wf1, Wh);
  gat_a12<<<dim3((NC * BN * 32) / 256), 256, 0, stream>>>(Wh, w1_1, b1_1, w2_1, b2_1, a1, a2);
  gat_colstats<<<dim3(N / 256, B), 256, 0, stream>>>(edges, a1, a2, colm, sinv);
  gat_msg<1><<<dim3(N / 64, B, NC), 128, 0, stream>>>(edges, a1, a2, colm, sinv, Wh, bias1, msgbuf);
  gat_elu_mean<<<dim3(BNH / 256), 256, 0, stream>>>(msgbuf, bias1, state2);

  gat_final<<<1, 256, 0, stream>>>(state2, W_out, b_out, (float*)d_out);
}